// Attention_31267361915369
// MI455X (gfx1250) — compile-verified
//
#include <hip/hip_runtime.h>
#include <math.h>

typedef __attribute__((ext_vector_type(2))) float v2f;
typedef __attribute__((ext_vector_type(8))) float v8f;

constexpr int Bc = 8;
constexpr int Tc = 4096;
constexpr int Dc = 1024;

__device__ __forceinline__ v8f wmma_f32(v2f a, v2f b, v8f c) {
    // D = A(16x4 f32) x B(4x16 f32) + C(16x16 f32)
    return __builtin_amdgcn_wmma_f32_16x16x4_f32(
        /*neg_a=*/false, a, /*neg_b=*/false, b,
        /*c_mod=*/(short)0, c, /*reuse_a=*/false, /*reuse_b=*/false);
}

// Branchless tanh: 1 - 2/(exp(2x)+1). Uses v_exp_f32 + v_rcp_f32 (TRANS ops,
// co-execute with WMMA). Saturates correctly for |x| large; ~1e-6 rel error.
__device__ __forceinline__ float tanh_fast(float v) {
    const float t = __expf(2.0f * v);
    return 1.0f - 2.0f * __builtin_amdgcn_rcpf(t + 1.0f);
}

// ---------------------------------------------------------------------------
// Kernel 1: s[b,t] = w2 . tanh(W1 x[b,t] + b1) + b2
// One block = 64 rows (four 16-row WMMA tiles) so W1 is streamed half as
// often. 8 waves split the 64 e-tiles; each wave processes e-tiles in PAIRS:
// inner k-loop = 6 loads -> 8 WMMAs (4 row tiles x 2 e-tiles).
// ---------------------------------------------------------------------------
__global__ __launch_bounds__(256) void score_kernel(
    const float* __restrict__ x, const float* __restrict__ W1,
    const float* __restrict__ b1, const float* __restrict__ w2,
    const float* __restrict__ b2, float* __restrict__ s)
{
    const int r0   = blockIdx.x * 64;       // flattened row base (b*T + t)
    const int lane = threadIdx.x & 31;
    const int wave = threadIdx.x >> 5;
    const int nl   = lane & 15;             // M (for A) / N (for B) index
    const int klo  = (lane >> 4) << 1;      // lanes 16-31 hold K=2,3

    __shared__ float sred[64];
    if (threadIdx.x < 64) sred[threadIdx.x] = 0.0f;
    __syncthreads();

    const float* xb = x + (size_t)(r0 + nl) * Dc;   // row tile rt at +rt*16*Dc

    float acc[4][8];
#pragma unroll
    for (int rt = 0; rt < 4; ++rt)
#pragma unroll
        for (int i = 0; i < 8; ++i) acc[rt][i] = 0.0f;

    // each wave owns 8 consecutive e-tiles, processed as 4 pairs
    for (int ep = 0; ep < 4; ++ep) {
        const int e0 = (wave * 8 + ep * 2) * 16;
        const int e1 = e0 + 16;
        const float* wb = W1 + (size_t)(e0 + nl) * Dc;   // e1 tile at +16*Dc

        v8f c[4][2] = {};
        for (int k = 0; k < Dc; k += 4) {
            const v2f b0 = *(const v2f*)(wb + k + klo);
            const v2f b1v = *(const v2f*)(wb + 16 * Dc + k + klo);
#pragma unroll
            for (int rt = 0; rt < 4; ++rt) {
                const v2f a = *(const v2f*)(xb + rt * 16 * Dc + k + klo);
                c[rt][0] = wmma_f32(a, b0,  c[rt][0]);
                c[rt][1] = wmma_f32(a, b1v, c[rt][1]);
            }
        }
        const float bb0 = b1[e0 + nl], ww0 = w2[e0 + nl];
        const float bb1 = b1[e1 + nl], ww1 = w2[e1 + nl];
#pragma unroll
        for (int rt = 0; rt < 4; ++rt)
#pragma unroll
            for (int i = 0; i < 8; ++i)
                acc[rt][i] += tanh_fast(c[rt][0][i] + bb0) * ww0
                            + tanh_fast(c[rt][1][i] + bb1) * ww1;
    }

    // reduce over the 16 columns (lanes) within each half-wave
#pragma unroll
    for (int rt = 0; rt < 4; ++rt) {
#pragma unroll
        for (int i = 0; i < 8; ++i) {
            float v = acc[rt][i];
            v += __shfl_xor(v, 1, 32);
            v += __shfl_xor(v, 2, 32);
            v += __shfl_xor(v, 4, 32);
            v += __shfl_xor(v, 8, 32);
            acc[rt][i] = v;
        }
    }
    if (nl == 0) {                        // lanes 0 (rows i) and 16 (rows i+8)
        const int rbase = (lane >> 4) * 8;
#pragma unroll
        for (int rt = 0; rt < 4; ++rt)
#pragma unroll
            for (int i = 0; i < 8; ++i)
                atomicAdd(&sred[rt * 16 + rbase + i], acc[rt][i]);
    }
    __syncthreads();
    if (threadIdx.x < 64) s[r0 + threadIdx.x] = sred[threadIdx.x] + b2[0];
}

// ---------------------------------------------------------------------------
// Kernel 2: per-batch max, e = exp(s-m), den = inclusive cumsum(e).
// One block per batch; 256 threads x 16 contiguous elements.
// ---------------------------------------------------------------------------
__global__ __launch_bounds__(256) void softmax_scan_kernel(
    const float* __restrict__ s, float* __restrict__ e, float* __restrict__ den)
{
    const int b   = blockIdx.x;
    const int tid = threadIdx.x;
    const float* sb = s + (size_t)b * Tc;
    float*       eb = e + (size_t)b * Tc;
    float*       db = den + (size_t)b * Tc;

    __shared__ float sm[256];

    float vals[16];
    float mx = -INFINITY;
#pragma unroll
    for (int i = 0; i < 16; ++i) {
        vals[i] = sb[tid * 16 + i];
        mx = fmaxf(mx, vals[i]);
    }
    sm[tid] = mx;
    __syncthreads();
    for (int off = 128; off > 0; off >>= 1) {
        if (tid < off) sm[tid] = fmaxf(sm[tid], sm[tid + off]);
        __syncthreads();
    }
    const float m = sm[0];
    __syncthreads();

    // exp + local inclusive prefix (store e as we go)
    float run = 0.0f;
#pragma unroll
    for (int i = 0; i < 16; ++i) {
        const float ev = __expf(vals[i] - m);
        eb[tid * 16 + i] = ev;
        run += ev;
        vals[i] = run;                    // local inclusive prefix
    }

    // block-wide inclusive scan of per-thread totals (Hillis-Steele)
    sm[tid] = run;
    __syncthreads();
    for (int off = 1; off < 256; off <<= 1) {
        const float mine = sm[tid];
        const float prev = (tid >= off) ? sm[tid - off] : 0.0f;
        __syncthreads();
        sm[tid] = mine + prev;
        __syncthreads();
    }
    const float excl = sm[tid] - run;
#pragma unroll
    for (int i = 0; i < 16; ++i) db[tid * 16 + i] = excl + vals[i];
}

// ---------------------------------------------------------------------------
// Kernel 3: out = tanh( cumsum_t(e * x@Wl^T)/den + x@Wr^T + bc )
// One block per (batch, 32-column d-tile) -> 256 blocks, halving x traffic.
// 8 waves split K=1024. Each scan step covers 32 rows x 32 cols:
// inner k-loop = 6 loads -> 8 WMMAs. Cross-wave reduce into DOUBLE-BUFFERED
// LDS with ONE barrier per step, so wave 0's scan/tanh/store epilogue for
// step tt overlaps the other waves' GEMM for step tt+1.
// ---------------------------------------------------------------------------
__global__ __launch_bounds__(256) void ctx_out_kernel(
    const float* __restrict__ x, const float* __restrict__ Wc,
    const float* __restrict__ bc, const float* __restrict__ e,
    const float* __restrict__ den, float* __restrict__ out)
{
    const int b    = blockIdx.x >> 5;        // / (D/32)
    const int dt   = blockIdx.x & 31;
    const int d0   = dt * 32;
    const int lane = threadIdx.x & 31;
    const int wave = threadIdx.x >> 5;
    const int nl   = lane & 15;
    const int hi   = lane >> 4;
    const int klo  = hi * 2;
    const int kbeg = wave * (Dc / 8);
    const int kend = kbeg + (Dc / 8);

    // double-buffered reduction: [buf][matrix][vgpr][lane]
    // matrix m: L side = rt*2 + j, R side = 4 + rt*2 + j  (rt=row tile, j=n tile)
    __shared__ float red[2][8][8][32];
    for (int i = threadIdx.x; i < 2 * 8 * 8 * 32; i += 256) ((float*)red)[i] = 0.0f;
    __syncthreads();

    // Wc base for n-tile 0: rows d0+nl (L) / +Dc (R); n-tile 1 at +32*Dc.
    const float* WcB = Wc + (size_t)(d0 + nl) * (2 * Dc);
    const float  bcv0 = bc[d0 + nl];
    const float  bcv1 = bc[d0 + 16 + nl];
    float carry[2] = {0.0f, 0.0f};

    for (int tt = 0; tt < Tc / 32; ++tt) {
        const int    t0 = tt * 32;
        const size_t r0 = (size_t)b * Tc + t0;
        const float* xb = x + (r0 + nl) * Dc;       // row tile 1 at +16*Dc

        v8f c[8] = {};
        for (int k = kbeg; k < kend; k += 4) {
            const v2f a0  = *(const v2f*)(xb + k + klo);
            const v2f a1  = *(const v2f*)(xb + 16 * Dc + k + klo);
            const v2f bl0 = *(const v2f*)(WcB + k + klo);
            const v2f bl1 = *(const v2f*)(WcB + 32 * Dc + k + klo);
            const v2f br0 = *(const v2f*)(WcB + Dc + k + klo);
            const v2f br1 = *(const v2f*)(WcB + 33 * Dc + k + klo);
            c[0] = wmma_f32(a0, bl0, c[0]);   // L rt0 n0
            c[1] = wmma_f32(a0, bl1, c[1]);   // L rt0 n1
            c[2] = wmma_f32(a1, bl0, c[2]);   // L rt1 n0
            c[3] = wmma_f32(a1, bl1, c[3]);   // L rt1 n1
            c[4] = wmma_f32(a0, br0, c[4]);   // R rt0 n0
            c[5] = wmma_f32(a0, br1, c[5]);   // R rt0 n1
            c[6] = wmma_f32(a1, br0, c[6]);   // R rt1 n0
            c[7] = wmma_f32(a1, br1, c[7]);   // R rt1 n1
        }
        const int buf = tt & 1;
#pragma unroll
        for (int m = 0; m < 8; ++m)
#pragma unroll
            for (int i = 0; i < 8; ++i)
                atomicAdd(&red[buf][m][i][lane], c[m][i]);
        __syncthreads();   // single barrier per step (double-buffered)

        if (wave == 0) {
            float zz[2][2][8], rr[2][2][8];   // [n tile][row tile][vgpr]
#pragma unroll
            for (int j = 0; j < 2; ++j)
#pragma unroll
                for (int rt = 0; rt < 2; ++rt)
#pragma unroll
                    for (int i = 0; i < 8; ++i) {
                        zz[j][rt][i] = red[buf][rt * 2 + j][i][lane];
                        rr[j][rt][i] = red[buf][4 + rt * 2 + j][i][lane];
                    }
            const float* ebp = e   + (size_t)b * Tc + t0 + hi * 8;
            const float* dbp = den + (size_t)b * Tc + t0 + hi * 8;
#pragma unroll
            for (int j = 0; j < 2; ++j)
#pragma unroll
                for (int rt = 0; rt < 2; ++rt)
#pragma unroll
                    for (int i = 0; i < 8; ++i)
                        zz[j][rt][i] *= ebp[rt * 16 + i];

            // prefix scan over 32 rows, independently per n-tile j
#pragma unroll
            for (int j = 0; j < 2; ++j) {
#pragma unroll
                for (int i = 1; i < 8; ++i) {
                    zz[j][0][i] += zz[j][0][i - 1];
                    zz[j][1][i] += zz[j][1][i - 1];
                }
                const float l0 = __shfl_xor(zz[j][0][7], 16, 32);
                const float l1 = __shfl_xor(zz[j][1][7], 16, 32);
                if (hi) {
#pragma unroll
                    for (int i = 0; i < 8; ++i) { zz[j][0][i] += l0; zz[j][1][i] += l1; }
                }
                const float sh0  = __shfl_xor(zz[j][0][7], 16, 32);
                const float tot0 = hi ? zz[j][0][7] : sh0;
                const float cj   = carry[j];
#pragma unroll
                for (int i = 0; i < 8; ++i) {
                    zz[j][0][i] += cj;
                    zz[j][1][i] += cj + tot0;
                }
                const float sh1 = __shfl_xor(zz[j][1][7], 16, 32);
                carry[j] = hi ? zz[j][1][7] : sh1;
            }

#pragma unroll
            for (int rt = 0; rt < 2; ++rt)
#pragma unroll
                for (int i = 0; i < 8; ++i) {
                    const float dinv = dbp[rt * 16 + i];
                    const size_t ro  = (r0 + rt * 16 + i + hi * 8) * Dc + d0 + nl;
                    out[ro]      = tanh_fast(zz[0][rt][i] / dinv + rr[0][rt][i] + bcv0);
                    out[ro + 16] = tanh_fast(zz[1][rt][i] / dinv + rr[1][rt][i] + bcv1);
                }
            // re-zero this buffer; other waves only redeposit here after the
            // NEXT barrier, which wave 0 reaches after this point.
#pragma unroll
            for (int m = 0; m < 8; ++m)
#pragma unroll
                for (int i = 0; i < 8; ++i)
                    red[buf][m][i][lane] = 0.0f;
        }
    }
}

extern "C" void kernel_launch(void* const* d_in, const int* in_sizes, int n_in,
                              void* d_out, int out_size, void* d_ws, size_t ws_size,
                              hipStream_t stream) {
    const float* x  = (const float*)d_in[0];
    const float* W1 = (const float*)d_in[1];
    const float* b1 = (const float*)d_in[2];
    const float* w2 = (const float*)d_in[3];
    const float* b2 = (const float*)d_in[4];
    const float* Wc = (const float*)d_in[5];
    const float* bc = (const float*)d_in[6];
    float* out = (float*)d_out;

    float* s   = (float*)d_ws;            // [B*T]
    float* e   = s + Bc * Tc;             // [B*T]
    float* den = e + Bc * Tc;             // [B*T]

    score_kernel<<<Bc * Tc / 64, 256, 0, stream>>>(x, W1, b1, w2, b2, s);
    softmax_scan_kernel<<<Bc, 256, 0, stream>>>(s, e, den);
    ctx_out_kernel<<<Bc * (Dc / 32), 256, 0, stream>>>(x, Wc, bc, e, den, out);
}